// MessagePassing_22239340659448
// MI455X (gfx1250) — compile-verified
//
#include <hip/hip_runtime.h>
#include <stdint.h>

// out[dst_e, f] += w_e * x[src_e, f]   (E edges, F=64 features, f32)
//
// MI455X design notes:
//  - Atomic-bound: E*F = 64M f32 atomics; out (25.6MB) lives in the 192MB L2,
//    so use no-return global_atomic_add_f32 scope:SCOPE_DEV (STOREcnt,
//    fire-and-forget, RMW resolved at device-coherent L2).
//  - Gather-bound second: one wave per edge, lane l loads float2 at
//    x[src*64 + 2l] -> two fully-coalesced 128B lines per row; x (25.6MB)
//    stays L2-resident and each row is reused ~10x (E/N), so default RT
//    temporal hints are correct.
//  - Edge stream (src,dst,w) staged into LDS with the CDNA5 async-copy path
//    (global_load_async_to_lds_b32, ASYNCcnt, s_wait_asynccnt), double-buffered
//    so the next chunk's DMA overlaps this chunk's gather+atomic work.
//  - WMMA is deliberately NOT used: 64 MFLOP total and a data-dependent
//    scatter reduction; no dense K axis for the matrix pipes.

#define FEAT   64
#define CHUNK  256
#define BLOCK  256
#define NWAVES 8

// Low 32 bits of a generic (flat) shared-memory address are the LDS byte
// offset (aperture encoding, CDNA5 ISA 10.2).
__device__ __forceinline__ unsigned lds_off(const void* p) {
    return (unsigned)(uintptr_t)p;
}

__global__ __launch_bounds__(BLOCK) void mp_scatter_kernel(
    const float* __restrict__ x,       // [N, 64]
    const float* __restrict__ ew,      // [E]
    const int*   __restrict__ ei,      // [2, E] row0=src, row1=dst
    float*       __restrict__ out,     // [N, 64], pre-zeroed
    int nEdges, int numChunks)
{
    __shared__ int   s_src[2][CHUNK];
    __shared__ int   s_dst[2][CHUNK];
    __shared__ float s_w  [2][CHUNK];

    const int t    = threadIdx.x;
    const int wave = t >> 5;
    const int lane = t & 31;

    const uint64_t srcBase = (uint64_t)(uintptr_t)ei;
    const uint64_t dstBase = (uint64_t)(uintptr_t)(ei + nEdges);
    const uint64_t wBase   = (uint64_t)(uintptr_t)ew;

    // Stage one 256-edge chunk (thread t stages edge slot t) via async DMA
    // into LDS. 3 async b32 transfers per wave per chunk.
    auto stage = [&](int chunk, int buf) {
        int e = chunk * CHUNK + t;
        if (e > nEdges - 1) e = nEdges - 1;           // clamp tail (harmless dup)
        uint64_t ga_src = srcBase + (uint64_t)e * 4u;
        uint64_t ga_dst = dstBase + (uint64_t)e * 4u;
        uint64_t ga_w   = wBase   + (uint64_t)e * 4u;
        unsigned l_src = lds_off(&s_src[buf][t]);
        unsigned l_dst = lds_off(&s_dst[buf][t]);
        unsigned l_w   = lds_off(&s_w  [buf][t]);
        asm volatile("global_load_async_to_lds_b32 %0, %1, off"
                     :: "v"(l_src), "v"(ga_src) : "memory");
        asm volatile("global_load_async_to_lds_b32 %0, %1, off"
                     :: "v"(l_dst), "v"(ga_dst) : "memory");
        asm volatile("global_load_async_to_lds_b32 %0, %1, off"
                     :: "v"(l_w),   "v"(ga_w)   : "memory");
    };

    int c = blockIdx.x;
    if (c >= numChunks) return;

    const int stride = gridDim.x;
    int buf = 0;
    stage(c, buf);                                    // prologue DMA

    for (; c < numChunks; c += stride, buf ^= 1) {
        // My async writes for chunk c are done; barrier makes every wave's
        // staged data visible block-wide (and closes out reads of buf^1).
        asm volatile("s_wait_asynccnt 0x0" ::: "memory");
        __syncthreads();

        // Kick off next chunk's DMA into the other buffer; it overlaps all
        // of the gather + atomic work below.
        if (c + stride < numChunks) stage(c + stride, buf ^ 1);

        int cnt = nEdges - c * CHUNK;
        if (cnt > CHUNK) cnt = CHUNK;

        // One edge per wave slot; 32 lanes x float2 = 64 features.
        // Atomics carry no memory clobber, so unrolled iterations let the
        // compiler hoist 4 slots' LDS + row gathers ahead of the atomics.
        #pragma unroll 4
        for (int slot = wave; slot < cnt; slot += NWAVES) {
            int   src = s_src[buf][slot];             // LDS broadcast reads
            int   dst = s_dst[buf][slot];
            float w   = s_w  [buf][slot];

            const float2 xv =
                *(const float2*)(x + (size_t)src * FEAT + lane * 2);
            float m0 = w * xv.x;
            float m1 = w * xv.y;

            uint64_t oa =
                (uint64_t)(uintptr_t)(out + (size_t)dst * FEAT + lane * 2);
            // No-return f32 atomics at device scope: RMW at L2 (coherent
            // across WGPs), STOREcnt-tracked, no result VGPR, no waits;
            // hardware stalls issue only if STOREcnt would overflow.
            asm volatile("global_atomic_add_f32 %0, %1, off scope:SCOPE_DEV"
                         :: "v"(oa), "v"(m0));
            asm volatile("global_atomic_add_f32 %0, %1, off offset:4 scope:SCOPE_DEV"
                         :: "v"(oa), "v"(m1));
        }
    }
}

extern "C" void kernel_launch(void* const* d_in, const int* in_sizes, int n_in,
                              void* d_out, int out_size, void* d_ws, size_t ws_size,
                              hipStream_t stream) {
    const float* x  = (const float*)d_in[0];
    const float* ew = (const float*)d_in[1];
    const int*   ei = (const int*)d_in[2];     // [2, E] int32 per harness mapping
    float* out = (float*)d_out;

    const int E = in_sizes[1];                 // edge_weight count
    const int numChunks = (E + CHUNK - 1) / CHUNK;

    // segment_sum starts from zeros; harness poisons d_out, so clear it.
    hipMemsetAsync(d_out, 0, (size_t)out_size * sizeof(float), stream);

    int grid = numChunks < 1024 ? numChunks : 1024;
    mp_scatter_kernel<<<grid, BLOCK, 0, stream>>>(x, ew, ei, out, E, numChunks);
}